// ResGCNBlock_38603166057035
// MI455X (gfx1250) — compile-verified
//
#include <hip/hip_runtime.h>
#include <hip/hip_bf16.h>

#define DFEAT 64
#define EPS_GN 1e-5f

typedef __attribute__((ext_vector_type(2))) float v2f;
typedef __attribute__((ext_vector_type(8))) float v8f;

// ---------------------------------------------------------------------------
// Zero-fill scratch (agg, deg, colsum, colsumsq) each launch: deterministic.
// ---------------------------------------------------------------------------
__global__ void zero_ws_kernel(float* __restrict__ p, long long n) {
    long long i = (long long)blockIdx.x * blockDim.x + threadIdx.x;
    long long stride = (long long)gridDim.x * blockDim.x;
    for (; i < n; i += stride) p[i] = 0.0f;
}

// ---------------------------------------------------------------------------
// h = x @ W  via V_WMMA_F32_16X16X4_F32 (full f32 precision).
// One wave -> 16 rows x 64 cols (4 accumulator tiles), K stepped by 4.
// W is staged in LDS PRE-PAIRED along K: lWp[p*64+n] = {W[2p][n], W[2p+1][n]},
// so every B fragment is a single aligned ds_load_b64 already in WMMA register
// layout (no cross-register packing movs in the inner loop).
// A 16x4 f32 fragment: lanes 0-15 hold M=lane, K=(0,1); lanes 16-31 K=(2,3).
// C/D 16x16 f32: VGPR v holds M = v + (lane>>4)*8, N = lane&15.
// ---------------------------------------------------------------------------
__global__ __launch_bounds__(256) void gemm_xw_wmma(
    const float* __restrict__ x, const float* __restrict__ W,
    float* __restrict__ h, int nnodes) {
    __shared__ float2 lWp[(DFEAT / 2) * DFEAT];  // 32 K-pairs x 64 cols = 16 KB
    for (int idx = threadIdx.x; idx < (DFEAT / 2) * DFEAT; idx += 256) {
        int p = idx >> 6;        // K-pair index
        int n = idx & 63;        // column
        float2 w;
        w.x = W[(2 * p) * DFEAT + n];
        w.y = W[(2 * p + 1) * DFEAT + n];
        lWp[idx] = w;
    }
    __syncthreads();

    const int lane = threadIdx.x & 31;
    const int wave = threadIdx.x >> 5;
    const int m0 = (blockIdx.x * 8 + wave) * 16;
    if (m0 >= nnodes) return;  // wave-uniform: EXEC stays all-ones for WMMA

    const int mrow = m0 + (lane & 15);
    const int kp_off = lane >> 4;  // which K-pair of the 4-wide K step
    const int ncol = lane & 15;

    v8f c0 = {}, c1 = {}, c2 = {}, c3 = {};
    const float* xrow = x + (size_t)mrow * DFEAT;

#pragma unroll
    for (int ks = 0; ks < DFEAT / 4; ++ks) {   // K step of 4: k = ks*4
        const int kp = ks * 2 + kp_off;        // this half-wave's K-pair
        // A fragment: {x[m][2kp], x[m][2kp+1]} -- one aligned b64 load
        float2 af = *(const float2*)(xrow + kp * 2);
        v2f a; a.x = af.x; a.y = af.y;

        const float2* wp = &lWp[kp * DFEAT + ncol];
        float2 f0 = wp[0];
        float2 f1 = wp[16];
        float2 f2 = wp[32];
        float2 f3 = wp[48];
        v2f b0, b1, b2, b3;
        b0.x = f0.x; b0.y = f0.y;
        b1.x = f1.x; b1.y = f1.y;
        b2.x = f2.x; b2.y = f2.y;
        b3.x = f3.x; b3.y = f3.y;

        c0 = __builtin_amdgcn_wmma_f32_16x16x4_f32(false, a, false, b0, (short)0, c0, false, false);
        c1 = __builtin_amdgcn_wmma_f32_16x16x4_f32(false, a, false, b1, (short)0, c1, false, false);
        c2 = __builtin_amdgcn_wmma_f32_16x16x4_f32(false, a, false, b2, (short)0, c2, false, false);
        c3 = __builtin_amdgcn_wmma_f32_16x16x4_f32(false, a, false, b3, (short)0, c3, false, false);
    }

#pragma unroll
    for (int v = 0; v < 8; ++v) {
        int row = m0 + (lane >> 4) * 8 + v;
        float* hp = h + (size_t)row * DFEAT + ncol;
        hp[0]  = c0[v];
        hp[16] = c1[v];
        hp[32] = c2[v];
        hp[48] = c3[v];
    }
}

// ---------------------------------------------------------------------------
// deg[dst] += 1 over all edges (float atomics; L2-resident, 0.4 MB array).
// ---------------------------------------------------------------------------
__global__ void degree_kernel(const long long* __restrict__ ei,
                              float* __restrict__ deg, int nedges) {
    long long i = (long long)blockIdx.x * blockDim.x + threadIdx.x;
    long long stride = (long long)gridDim.x * blockDim.x;
    for (; i < nedges; i += stride) {
        int dst = (int)ei[(long long)nedges + i];
        atomicAdd(&deg[dst], 1.0f);
    }
}

// ---------------------------------------------------------------------------
// agg[dst] += norm * h[src], 64 threads per edge (feature = tid&63).
// h and agg both fit in the 192 MB L2 -> random traffic stays on-chip.
// ---------------------------------------------------------------------------
__global__ __launch_bounds__(256) void edge_agg_kernel(
    const long long* __restrict__ ei, const float* __restrict__ h,
    const float* __restrict__ deg, float* __restrict__ agg, int nedges) {
    int f = threadIdx.x & 63;
    long long e = (long long)blockIdx.x * 4 + (threadIdx.x >> 6);
    if (e >= nedges) return;
    int src = (int)ei[e];
    int dst = (int)ei[(long long)nedges + e];
    float norm = rsqrtf(deg[src] + 1.0f) * rsqrtf(deg[dst] + 1.0f);
    float val = h[(size_t)src * DFEAT + f] * norm;
    atomicAdd(&agg[(size_t)dst * DFEAT + f], val);
}

// ---------------------------------------------------------------------------
// agg += h * dinv^2 + b, and accumulate per-feature sum / sum-of-squares
// (LDS tree reduction -> 128 global atomics per block of 64 rows).
// ---------------------------------------------------------------------------
__global__ __launch_bounds__(256) void selfloop_stats_kernel(
    const float* __restrict__ h, const float* __restrict__ deg,
    const float* __restrict__ b, float* __restrict__ agg,
    float* __restrict__ cs, float* __restrict__ css, int nnodes) {
    __shared__ float s_sum[256];
    __shared__ float s_sq[256];
    int f = threadIdx.x & 63;
    int sub = threadIdx.x >> 6;  // 0..3
    int base = blockIdx.x * 64;
    float bf = b[f];
    float psum = 0.0f, psq = 0.0f;
    for (int r = sub; r < 64; r += 4) {
        int i = base + r;
        if (i < nnodes) {
            float dinv2 = 1.0f / (deg[i] + 1.0f);
            size_t off = (size_t)i * DFEAT + f;
            float v = agg[off] + h[off] * dinv2 + bf;
            agg[off] = v;
            psum += v;
            psq += v * v;
        }
    }
    s_sum[threadIdx.x] = psum;
    s_sq[threadIdx.x] = psq;
    __syncthreads();
    if (threadIdx.x < 64) {
        float t  = s_sum[f] + s_sum[f + 64] + s_sum[f + 128] + s_sum[f + 192];
        float t2 = s_sq[f]  + s_sq[f + 64]  + s_sq[f + 128]  + s_sq[f + 192];
        atomicAdd(&cs[f], t);
        atomicAdd(&css[f], t2);
    }
}

// ---------------------------------------------------------------------------
// GraphNorm + ReLU + residual. var = E[a^2] - mu^2*(2*alpha - alpha^2).
// ---------------------------------------------------------------------------
__global__ void finalize_kernel(
    const float* __restrict__ agg, const float* __restrict__ x,
    const float* __restrict__ cs, const float* __restrict__ css,
    const float* __restrict__ gamma, const float* __restrict__ beta,
    const float* __restrict__ alpha, float* __restrict__ out, int nnodes) {
    long long idx = (long long)blockIdx.x * blockDim.x + threadIdx.x;
    long long total = (long long)nnodes * DFEAT;
    long long stride = (long long)gridDim.x * blockDim.x;
    float invN = 1.0f / (float)nnodes;
    for (; idx < total; idx += stride) {
        int f = (int)(idx & 63);
        float mu = cs[f] * invN;
        float m2 = css[f] * invN;
        float al = alpha[f];
        float var = m2 - mu * mu * (2.0f * al - al * al);
        float sub = agg[idx] - al * mu;
        float o = gamma[f] * sub * rsqrtf(var + EPS_GN) + beta[f];
        o = fmaxf(o, 0.0f) + x[idx];
        out[idx] = o;
    }
}

// ---------------------------------------------------------------------------
extern "C" void kernel_launch(void* const* d_in, const int* in_sizes, int n_in,
                              void* d_out, int out_size, void* d_ws, size_t ws_size,
                              hipStream_t stream) {
    const float* x          = (const float*)d_in[0];
    const long long* ei     = (const long long*)d_in[1];  // int64 edge_index [2, E]
    const float* W          = (const float*)d_in[2];
    const float* b          = (const float*)d_in[3];
    const float* gamma      = (const float*)d_in[4];
    const float* beta       = (const float*)d_in[5];
    const float* alpha      = (const float*)d_in[6];
    float* out              = (float*)d_out;

    const int nnodes = in_sizes[0] / DFEAT;
    const int nedges = in_sizes[1] / 2;

    // Workspace layout: h | agg | deg | colsum | colsumsq
    float* ws  = (float*)d_ws;
    float* h   = ws;
    float* agg = h + (size_t)nnodes * DFEAT;
    float* deg = agg + (size_t)nnodes * DFEAT;
    float* cs  = deg + nnodes;
    float* css = cs + DFEAT;

    // 1. Zero agg + deg + column stats (everything after h).
    long long zcount = (long long)nnodes * DFEAT + nnodes + 2 * DFEAT;
    zero_ws_kernel<<<2048, 256, 0, stream>>>(agg, zcount);

    // 2. h = x @ W (WMMA f32).
    int mtiles = (nnodes + 15) / 16;
    int gemm_blocks = (mtiles + 7) / 8;
    gemm_xw_wmma<<<gemm_blocks, 256, 0, stream>>>(x, W, h, nnodes);

    // 3. Degrees.
    degree_kernel<<<2048, 256, 0, stream>>>(ei, deg, nedges);

    // 4. Edge aggregation (4 edges per 256-thread block).
    int agg_blocks = (nedges + 3) / 4;
    edge_agg_kernel<<<agg_blocks, 256, 0, stream>>>(ei, h, deg, agg, nedges);

    // 5. Self-loop + bias + column statistics (64 rows per block).
    int stat_blocks = (nnodes + 63) / 64;
    selfloop_stats_kernel<<<stat_blocks, 256, 0, stream>>>(h, deg, b, agg, cs, css, nnodes);

    // 6. GraphNorm + ReLU + residual.
    finalize_kernel<<<4096, 256, 0, stream>>>(agg, x, cs, css, gamma, beta, alpha, out, nnodes);
}